// BigramModel_30837865185904
// MI455X (gfx1250) — compile-verified
//
#include <hip/hip_runtime.h>
#include <math.h>

typedef __attribute__((ext_vector_type(2))) float v2f;
typedef __attribute__((ext_vector_type(8))) float v8f;

// Problem constants (from the reference)
constexpr int VOCAB = 100277;
constexpr int EMBED = 32;
constexpr int NROWS = 4096;      // B * S = 8 * 512
constexpr int WAVES = 8;         // 256 threads / wave32
constexpr int NITER = 16;        // 16x16 tiles per wave
constexpr int WAVE_COLS = 16 * NITER;              // 256
constexpr int WG_COLS   = WAVES * WAVE_COLS;       // 2048
constexpr int NTILES    = (VOCAB + WG_COLS - 1) / WG_COLS;  // 49

constexpr float L2E = 1.4426950408889634f;  // log2(e)
constexpr float LN2 = 0.6931471805599453f;  // ln(2)

// Raw v_exp_f32: args here are always <= 0; below-denormal range flushes to 0,
// which is exactly the desired contribution -- skip libm's range fixup.
__device__ __forceinline__ float fast_exp2(float x) {
    return __builtin_amdgcn_exp2f(x);
}

// Base-2 domain (max, sum) merge: m in log2 units, s = sum of 2^(y - m).
__device__ __forceinline__ void combine2(float& m, float& s, float m2, float s2) {
    float nm = fmaxf(m, m2);
    s = fmaf(s, fast_exp2(m - nm), s2 * fast_exp2(m2 - nm));
    m = nm;
}

// Online update with y = x * log2(e) already in base-2 domain.
__device__ __forceinline__ void online_update2(float& m, float& s, float y) {
    float nm = fmaxf(m, y);
    s = fmaf(s, fast_exp2(m - nm), fast_exp2(y - nm));
    m = nm;
}

// Kernel 1: fused GEMM (WMMA f32 16x16x4) + logits store + online per-row softmax partials.
__global__ __launch_bounds__(256)
void bigram_logits_kernel(const int* __restrict__ input_seq,
                          const float* __restrict__ emb,
                          const float* __restrict__ W,
                          const float* __restrict__ bias,
                          float* __restrict__ logits,
                          float* __restrict__ part_max,
                          float* __restrict__ part_sum)
{
    const int lane = threadIdx.x & 31;
    const int wv   = threadIdx.x >> 5;
    const int lr   = lane & 15;          // position within 16-lane half
    const int hi   = lane >> 4;          // 0 = lanes 0-15, 1 = lanes 16-31
    const int mbase = blockIdx.x * 16;
    const long long colbase = (long long)blockIdx.y * WG_COLS + (long long)wv * WAVE_COLS;

    // ---- A fragments: 16x32 f32, reused across all 16 N-iterations ----
    // A 16x4 layout: lanes 0-15 row=lr K={4k+0,4k+1}; lanes 16-31 row=lr K={4k+2,4k+3}
    const int tok = input_seq[mbase + lr];
    v2f a[8];
#pragma unroll
    for (int k = 0; k < 8; ++k)
        a[k] = *(const v2f*)(emb + (long long)tok * EMBED + 4 * k + 2 * hi);

    // per-lane running softmax state (base-2 domain) for this lane's 8 C/D rows
    float rmax[8], rsum[8];
#pragma unroll
    for (int r = 0; r < 8; ++r) { rmax[r] = -INFINITY; rsum[r] = 0.0f; }

    if (__builtin_expect(colbase + WAVE_COLS <= VOCAB, 1)) {
        // ================= FAST PATH (wave-uniform: every lane/col valid) ===========
        const float* wp = W + (colbase + lr) * EMBED + 2 * hi;  // lane's W row, k-pair base
        const float* bp = bias + colbase + lr;
        float* op[8];
#pragma unroll
        for (int r = 0; r < 8; ++r)
            op[r] = logits + (long long)(mbase + r + 8 * hi) * VOCAB + colbase + lr;

        for (int it = 0; it < NITER; ++it) {
            v8f c = {};
#pragma unroll
            for (int k = 0; k < 8; ++k) {
                v2f bf = *(const v2f*)(wp + it * 16 * EMBED + 4 * k);
                c = __builtin_amdgcn_wmma_f32_16x16x4_f32(
                        false, a[k], false, bf, (short)0, c, false, false);
            }
            const float bv = bp[it * 16];
#pragma unroll
            for (int r = 0; r < 8; ++r) {
                float x = c[r] + bv;                  // C/D VGPR r -> row r + 8*hi
                op[r][it * 16] = x;                   // plain store, immediate offset
                online_update2(rmax[r], rsum[r], x * L2E);
            }
        }
    } else {
        // ================= TAIL PATH (only the last wave of the last N-tile) ========
        for (int it = 0; it < NITER; ++it) {
            const long long v = colbase + it * 16 + lr;
            const bool valid = (v < VOCAB);
            const long long vc = valid ? v : (long long)(VOCAB - 1);

            v8f c = {};
#pragma unroll
            for (int k = 0; k < 8; ++k) {
                v2f bf = *(const v2f*)(W + vc * EMBED + 4 * k + 2 * hi);
                c = __builtin_amdgcn_wmma_f32_16x16x4_f32(
                        false, a[k], false, bf, (short)0, c, false, false);
            }
            const float bv = bias[vc];
#pragma unroll
            for (int r = 0; r < 8; ++r) {
                float x = c[r] + bv;
                if (valid) {
                    logits[(long long)(mbase + r + 8 * hi) * VOCAB + v] = x;
                    online_update2(rmax[r], rsum[r], x * L2E);
                }
            }
        }
    }

    // ---- reduce (max,sum) across the 16 lanes of each half (xor stays inside the half) ----
#pragma unroll
    for (int r = 0; r < 8; ++r) {
        float m = rmax[r], s = rsum[r];
#pragma unroll
        for (int off = 1; off < 16; off <<= 1) {
            float m2 = __shfl_xor(m, off, 32);
            float s2 = __shfl_xor(s, off, 32);
            combine2(m, s, m2, s2);
        }
        rmax[r] = m; rsum[r] = s;
    }

    // ---- cross-wave combine via LDS ----
    __shared__ float lmax[WAVES][16];
    __shared__ float lsum[WAVES][16];
    if (lr == 0) {
#pragma unroll
        for (int r = 0; r < 8; ++r) {
            lmax[wv][r + 8 * hi] = rmax[r];
            lsum[wv][r + 8 * hi] = rsum[r];
        }
    }
    __syncthreads();

    if (threadIdx.x < 16) {
        const int row = threadIdx.x;
        float m = lmax[0][row], s = lsum[0][row];
#pragma unroll
        for (int w = 1; w < WAVES; ++w)
            combine2(m, s, lmax[w][row], lsum[w][row]);
        const long long grow = mbase + row;
        part_max[grow * NTILES + blockIdx.y] = m;   // base-2 domain partials
        part_sum[grow * NTILES + blockIdx.y] = s;
    }
}

// Kernel 2: combine per-tile partials -> logsumexp, gather target logit, mean NLL.
__global__ __launch_bounds__(1024)
void bigram_loss_kernel(const int* __restrict__ predictions,
                        const float* __restrict__ logits,
                        const float* __restrict__ part_max,
                        const float* __restrict__ part_sum,
                        float* __restrict__ loss_out)
{
    __shared__ float red[32];
    float local = 0.0f;
    for (int row = threadIdx.x; row < NROWS; row += 1024) {
        float m = -INFINITY, s = 0.0f;
        for (int t = 0; t < NTILES; ++t)
            combine2(m, s, part_max[(long long)row * NTILES + t],
                           part_sum[(long long)row * NTILES + t]);
        // s >= 1 here (the max element contributes 2^0), so raw v_log_f32 is safe.
        const float lse = (m + __builtin_amdgcn_logf(s)) * LN2;
        const int tgt = predictions[row];
        const float lt = logits[(long long)row * VOCAB + tgt];
        local += (lse - lt);
    }
#pragma unroll
    for (int off = 16; off >= 1; off >>= 1)
        local += __shfl_xor(local, off, 32);
    const int lane = threadIdx.x & 31;
    const int wv = threadIdx.x >> 5;
    if (lane == 0) red[wv] = local;
    __syncthreads();
    if (threadIdx.x == 0) {
        float sum = 0.0f;
#pragma unroll
        for (int w = 0; w < 32; ++w) sum += red[w];
        loss_out[0] = sum / (float)NROWS;
    }
}

extern "C" void kernel_launch(void* const* d_in, const int* in_sizes, int n_in,
                              void* d_out, int out_size, void* d_ws, size_t ws_size,
                              hipStream_t stream) {
    const int*   input_seq   = (const int*)d_in[0];
    const int*   predictions = (const int*)d_in[1];
    const float* emb         = (const float*)d_in[2];
    const float* W           = (const float*)d_in[3];
    const float* bias        = (const float*)d_in[4];

    float* logits = (float*)d_out;
    float* loss   = logits + (long long)NROWS * VOCAB;

    float* part_max = (float*)d_ws;                       // NROWS*NTILES floats
    float* part_sum = part_max + (long long)NROWS * NTILES;

    dim3 grid(NROWS / 16, NTILES);
    bigram_logits_kernel<<<grid, 256, 0, stream>>>(input_seq, emb, W, bias,
                                                   logits, part_max, part_sum);
    bigram_loss_kernel<<<1, 1024, 0, stream>>>(predictions, logits,
                                               part_max, part_sum, loss);
}